// WhdrHingeLoss_15994458211236
// MI455X (gfx1250) — compile-verified
//
#include <hip/hip_runtime.h>
#include <stdint.h>
#include <math.h>

#define EPSF 1e-10f
#define WPB 8  // waves per block (wave32)

// ---- CDNA5 async global->LDS path (probe-confirmed on this toolchain) ------
#if defined(__AMDGCN__) && defined(__has_builtin)
#  if __has_builtin(__builtin_amdgcn_global_load_async_to_lds_b64) && \
      __has_builtin(__builtin_amdgcn_s_wait_asynccnt)
#    define USE_ASYNC_LDS 1
#  endif
#endif
#ifndef USE_ASYNC_LDS
#  define USE_ASYNC_LDS 0
#endif

#if USE_ASYNC_LDS
typedef int v2i __attribute__((ext_vector_type(2)));
typedef __attribute__((address_space(1))) v2i* gv2i_t;  // global AS
typedef __attribute__((address_space(3))) v2i* lv2i_t;  // LDS AS

__device__ __forceinline__ gv2i_t as_g(const void* p) {
  return (gv2i_t)(unsigned long long)(uintptr_t)p;
}
__device__ __forceinline__ lv2i_t as_l(void* p) {
  // generic LDS pointer: low 32 bits are the LDS offset
  return (lv2i_t)(unsigned int)(unsigned long long)(uintptr_t)p;
}
#endif

__device__ __forceinline__ int clampi(int v, int hi) {
  return v < 0 ? 0 : (v > hi ? hi : v);
}

// ---------------- Phase A: per-segment weight sums --------------------------
__global__ __launch_bounds__(256) void whdr_segw_kernel(
    const float* __restrict__ cmp, const int* __restrict__ numComp,
    float* __restrict__ segw, int B, int M, int S, int L) {
  const int lane = threadIdx.x & 31;
  const int wave = threadIdx.x >> 5;
  const int gw = blockIdx.x * WPB + wave;  // gw = b*S + s
  if (gw >= B * S) return;
  const int b = gw / S, s = gw - b * S;

  int nc = numComp[b];
  if (nc < 1) nc = 1;
  if (nc > M) nc = M;
  const int m0 = s * L;
  const int m1 = min(m0 + L, nc);

  const float* cb = cmp + (size_t)b * M * 6;
  float acc = 0.f;
  for (int m = m0 + lane; m < m1; m += 32)
    acc += __builtin_nontemporal_load(cb + (size_t)m * 6 + 5);
#pragma unroll
  for (int off = 16; off; off >>= 1) acc += __shfl_xor(acc, off, 32);
  if (lane == 0) segw[gw] = acc;
}

// ---------------- Phase B: per-segment affine reduction ---------------------
// Segment -> (A,B,P,Q): exit_loss = A*x + B, seg_total = P*x + Q,
// where x is the loss entering the segment. Incoming wsum is exact (phase A).
__global__ __launch_bounds__(256) void whdr_seg_kernel(
    const float* __restrict__ img, const float* __restrict__ cmp,
    const int* __restrict__ numComp, const float* __restrict__ segw,
    float* __restrict__ segA, float* __restrict__ segB,
    float* __restrict__ segP, float* __restrict__ segQ,
    int B, int Wdim, int M, int S, int L) {
  const int lane = threadIdx.x & 31;
  const int wave = threadIdx.x >> 5;
  const int gw = blockIdx.x * WPB + wave;  // gw = b*S + s
  if (gw >= B * S) return;                 // wave-uniform, no block syncs used
  const int b = gw / S, s = gw - b * S;

  int nc = numComp[b];
  if (nc < 1) nc = 1;
  if (nc > M) nc = M;
  const int mstart = s * L;
  const int mend = min(mstart + L, nc);

  // incoming weight-sum = sum of earlier segments' weight sums (S <= 32)
  float wb = (lane < s) ? segw[b * S + lane] : 0.f;
#pragma unroll
  for (int off = 16; off; off >>= 1) wb += __shfl_xor(wb, off, 32);

  float Aseg = 1.f, Bseg = 0.f, Pseg = 0.f, Qseg = 0.f;

#if USE_ASYNC_LDS
  __shared__ float sbuf[WPB][2][32 * 6];  // double-buffered 24B/lane rows
#endif

  if (mstart < mend) {
    const float border1 = 1.0f / 1.15f;  // 1/(1+DELTA+MARGIN)
    const float border2 = 1.15f;         // 1+DELTA+MARGIN
    const float br = 1.05f;              // 1+DELTA-MARGIN
    const float bl = 1.0f / 1.05f;

    const float* imgb = img + (size_t)b * Wdim * Wdim;
    const float* cb = cmp + (size_t)b * M * 6;
    float carry_wsum = wb;
    const int nchunks = (mend - mstart + 31) >> 5;

#if USE_ASYNC_LDS
    {  // issue chunk 0
      const int m0c = min(mstart + lane, mend - 1);
      const float* src = cb + (size_t)m0c * 6;
      void* dst = (void*)&sbuf[wave][0][lane * 6];
      __builtin_amdgcn_global_load_async_to_lds_b64(as_g(src), as_l(dst), 0, 0);
      __builtin_amdgcn_global_load_async_to_lds_b64(as_g(src), as_l(dst), 8, 0);
      __builtin_amdgcn_global_load_async_to_lds_b64(as_g(src), as_l(dst), 16, 0);
    }
    int buf = 0;
#endif

    for (int ci = 0; ci < nchunks; ++ci) {
      const int base = mstart + (ci << 5);
      const int m = base + lane;
      const bool valid = m < mend;

      float c0, c1, c2, c3, dk, w;
#if USE_ASYNC_LDS
      if (ci + 1 < nchunks) {  // stream next chunk into the other buffer
        const int mn = min(base + 32 + lane, mend - 1);
        const float* srcn = cb + (size_t)mn * 6;
        void* dstn = (void*)&sbuf[wave][buf ^ 1][lane * 6];
        __builtin_amdgcn_global_load_async_to_lds_b64(as_g(srcn), as_l(dstn), 0, 0);
        __builtin_amdgcn_global_load_async_to_lds_b64(as_g(srcn), as_l(dstn), 8, 0);
        __builtin_amdgcn_global_load_async_to_lds_b64(as_g(srcn), as_l(dstn), 16, 0);
        __builtin_amdgcn_s_wait_asynccnt(3);  // previous chunk's 3 ops done
      } else {
        __builtin_amdgcn_s_wait_asynccnt(0);
      }
      const float* row = &sbuf[wave][buf][lane * 6];
      c0 = row[0]; c1 = row[1]; c2 = row[2]; c3 = row[3]; dk = row[4]; w = row[5];
      buf ^= 1;
#else
      const int mm = valid ? m : mend - 1;
      const float* row = cb + (size_t)mm * 6;
      if (ci + 1 < nchunks) __builtin_prefetch(row + 32 * 6, 0, 3);
      c0 = __builtin_nontemporal_load(row + 0);
      c1 = __builtin_nontemporal_load(row + 1);
      c2 = __builtin_nontemporal_load(row + 2);
      c3 = __builtin_nontemporal_load(row + 3);
      dk = __builtin_nontemporal_load(row + 4);
      w  = __builtin_nontemporal_load(row + 5);
#endif

      // gather the two pixels (images stay L2-resident; NT stream protects L2)
      const int x1 = clampi((int)c0, Wdim - 1), y1 = clampi((int)c1, Wdim - 1);
      const int x2 = clampi((int)c2, Wdim - 1), y2 = clampi((int)c3, Wdim - 1);
      const float r1 = imgb[y1 * Wdim + x1];
      const float r2 = imgb[y2 * Wdim + x2];
      const float ratio = r1 / (r2 + EPSF);
      float h;
      if (dk == 1.0f)      h = fmaxf(ratio - border1, 0.f);
      else if (dk == 2.0f) h = fmaxf(border2 - ratio, 0.f);
      else                 h = fmaxf(ratio - br, 0.f) + fmaxf(bl - ratio, 0.f);

      // (1) lane prefix-sum of weights -> exact W_m (carry from phase A)
      const float wv = valid ? w : 0.f;
      float pw = wv;
#pragma unroll
      for (int d = 1; d < 32; d <<= 1) {
        float t = __shfl_up(pw, d, 32);
        if (lane >= d) pw += t;
      }
      const float Wl = carry_wsum + pw;

      // (2) per-lane affine map f(x) = (x + w*h)/W ; identity on invalid lanes
      float A, Bv;
      if (valid) {
        const float inv = 1.0f / Wl;
        A = inv;
        Bv = (wv * h) * inv;
      } else {
        A = 1.0f;
        Bv = 0.f;
      }
      // inclusive scan composing affine maps: new = cur ∘ prev
#pragma unroll
      for (int d = 1; d < 32; d <<= 1) {
        const float Ap = __shfl_up(A, d, 32);
        const float Bp = __shfl_up(Bv, d, 32);
        if (lane >= d) {
          Bv = fmaf(A, Bp, Bv);  // uses old A
          A *= Ap;
        }
      }

      // (3) chunk aggregates: lane-31 composite + per-lane sums over valid
      const float A31 = __shfl(A, 31, 32);
      const float B31 = __shfl(Bv, 31, 32);
      float Ac = valid ? A : 0.f;
      float Bc = valid ? Bv : 0.f;
#pragma unroll
      for (int off = 16; off; off >>= 1) {
        Ac += __shfl_xor(Ac, off, 32);
        Bc += __shfl_xor(Bc, off, 32);
      }

      // fold chunk into segment-level (A,B,P,Q) — order matters (old Bseg!)
      Pseg = fmaf(Ac, Aseg, Pseg);
      Qseg += fmaf(Ac, Bseg, Bc);
      Bseg = fmaf(A31, Bseg, B31);
      Aseg *= A31;

      carry_wsum = __shfl(Wl, 31, 32);
    }
  }

  if (lane == 0) {
    segA[gw] = Aseg; segB[gw] = Bseg; segP[gw] = Pseg; segQ[gw] = Qseg;
  }
}

// ------- Phase C: sequential segment combine per batch + final mean ---------
__global__ __launch_bounds__(256) void whdr_final_kernel(
    const float* __restrict__ segA, const float* __restrict__ segB,
    const float* __restrict__ segP, const float* __restrict__ segQ,
    float* __restrict__ out, int B, int S) {
  __shared__ float sh[256];
  const int t = threadIdx.x;
  float v = 0.f;
  for (int b = t; b < B; b += 256) {
    float x = 0.f, tot = 0.f;
    for (int s = 0; s < S; ++s) {
      const int i = b * S + s;
      tot += fmaf(segP[i], x, segQ[i]);
      x = fmaf(segA[i], x, segB[i]);
    }
    v += tot;
  }
  sh[t] = v;
  __syncthreads();
  for (int off = 128; off > 0; off >>= 1) {
    if (t < off) sh[t] += sh[t + off];
    __syncthreads();
  }
  if (t == 0) out[0] = sh[0] / (float)B;
}

extern "C" void kernel_launch(void* const* d_in, const int* in_sizes, int n_in,
                              void* d_out, int out_size, void* d_ws, size_t ws_size,
                              hipStream_t stream) {
  const float* img = (const float*)d_in[0];      // (B,1,H,W) f32
  const float* cmp = (const float*)d_in[1];      // (B,M,6)   f32
  const int* numComp = (const int*)d_in[2];      // (B,)      i32

  const int B = in_sizes[2];
  const int HW = in_sizes[0] / B;
  const int M = in_sizes[1] / (B * 6);
  const int Wdim = (int)lround(sqrt((double)HW));  // square images (256)

  // choose segment count (<=32 for lane-parallel prefix; shrink if ws tight)
  int S = 16;
  while (S > 1 && (size_t)(5 * B * S) * sizeof(float) > ws_size) S >>= 1;
  const int L = (((M + S - 1) / S) + 31) & ~31;  // 32-aligned segment length

  float* segw = (float*)d_ws;
  float* segA = segw + (size_t)B * S;
  float* segB = segA + (size_t)B * S;
  float* segP = segB + (size_t)B * S;
  float* segQ = segP + (size_t)B * S;
  float* out = (float*)d_out;

  const int nwaves = B * S;
  const int blocks = (nwaves + WPB - 1) / WPB;
  whdr_segw_kernel<<<blocks, WPB * 32, 0, stream>>>(cmp, numComp, segw, B, M, S, L);
  whdr_seg_kernel<<<blocks, WPB * 32, 0, stream>>>(img, cmp, numComp, segw,
                                                   segA, segB, segP, segQ,
                                                   B, Wdim, M, S, L);
  whdr_final_kernel<<<1, 256, 0, stream>>>(segA, segB, segP, segQ, out, B, S);
}